// GraphSAGE_22411139350784
// MI455X (gfx1250) — compile-verified
//
#include <hip/hip_runtime.h>
#include <math.h>

// ---------------------------------------------------------------------------
// GraphSAGE 2-layer inference for MI455X (gfx1250, wave32).
// GEMMs use V_WMMA_F32_16X16X4_F32 (f32 in/out, matches reference precision).
// Scatter phase is L2-resident atomic-add traffic (x and sum each 51MB < 192MB L2).
// ---------------------------------------------------------------------------

#define NN 100000
#define NE 1600000
#define DH 128
#define DO 40

typedef float v2f __attribute__((ext_vector_type(2)));
typedef float v8f __attribute__((ext_vector_type(8)));

// -------------------------- utility kernels --------------------------------

__global__ __launch_bounds__(256) void zero_kernel(float* __restrict__ p, long n) {
    long i = (long)blockIdx.x * 256 + threadIdx.x;
    if (i < n) p[i] = 0.0f;
}

// One wave (32 lanes) per edge; each lane atomically adds 4 of the 128 feats.
__global__ __launch_bounds__(256) void scatter_kernel(
    const float* __restrict__ feat,
    const long long* __restrict__ src,
    const long long* __restrict__ dst,
    float* __restrict__ acc,
    float* __restrict__ deg,   // nullptr => skip degree accumulation
    int nEdges)
{
    int e    = ((int)blockIdx.x * 256 + (int)threadIdx.x) >> 5;  // wave id = edge id
    int lane = threadIdx.x & 31;
    if (e >= nEdges) return;                                     // wave-uniform
    long long s = src[e];
    long long d = dst[e];
    const float* f = feat + (size_t)s * DH;
    float*       o = acc  + (size_t)d * DH;
#pragma unroll
    for (int i = 0; i < 4; ++i) {
        int c = lane + 32 * i;
        atomicAdd(&o[c], f[c]);
    }
    if (deg != nullptr && lane == 0) atomicAdd(&deg[d], 1.0f);
}

// 16-lane butterfly reductions (wave32: masks 1,2,4,8 stay within each half).
__device__ __forceinline__ float hadd16(float v) {
    v += __shfl_xor(v, 1);
    v += __shfl_xor(v, 2);
    v += __shfl_xor(v, 4);
    v += __shfl_xor(v, 8);
    return v;
}
__device__ __forceinline__ float hmax16(float v) {
    v = fmaxf(v, __shfl_xor(v, 1));
    v = fmaxf(v, __shfl_xor(v, 2));
    v = fmaxf(v, __shfl_xor(v, 4));
    v = fmaxf(v, __shfl_xor(v, 8));
    return v;
}

// ------------------------------ layer 1 ------------------------------------
// h = relu(l2norm(mean @ Wl + b + x @ Wr)), all 128-wide. One wave = 16 rows.
__global__ __launch_bounds__(256) void sage_layer1(
    const float* __restrict__ sum, const float* __restrict__ deg,
    const float* __restrict__ x,
    const float* __restrict__ Wl, const float* __restrict__ bl,
    const float* __restrict__ Wr,
    float* __restrict__ h)
{
    const int wave = threadIdx.x >> 5;
    const int lane = threadIdx.x & 31;
    const int m    = lane & 15;      // A-row / B-col / C-col within tile
    const int half = lane >> 4;      // K-group (A/B), M-group (C/D)
    const int row0 = ((int)blockIdx.x * 8 + wave) * 16;
    if (row0 >= NN) return;          // wave-uniform: EXEC stays all-ones for WMMA

    const int   row  = row0 + m;
    const float invd = 1.0f / fmaxf(deg[row], 1.0f);
    const float* aRow = sum + (size_t)row * DH;
    const float* xRow = x   + (size_t)row * DH;

    v8f acc[8] = {};

    // mean @ Wl
    for (int k0 = 0; k0 < DH; k0 += 4) {
        const int k = k0 + half * 2;
        v2f a;
        a.x = aRow[k]     * invd;
        a.y = aRow[k + 1] * invd;
#pragma unroll
        for (int nt = 0; nt < 8; ++nt) {
            v2f b;
            b.x = Wl[(size_t)k       * DH + nt * 16 + m];
            b.y = Wl[(size_t)(k + 1) * DH + nt * 16 + m];
            acc[nt] = __builtin_amdgcn_wmma_f32_16x16x4_f32(
                false, a, false, b, (short)0, acc[nt], false, false);
        }
    }
    // x @ Wr
    for (int k0 = 0; k0 < DH; k0 += 4) {
        const int k = k0 + half * 2;
        v2f a;
        a.x = xRow[k];
        a.y = xRow[k + 1];
#pragma unroll
        for (int nt = 0; nt < 8; ++nt) {
            v2f b;
            b.x = Wr[(size_t)k       * DH + nt * 16 + m];
            b.y = Wr[(size_t)(k + 1) * DH + nt * 16 + m];
            acc[nt] = __builtin_amdgcn_wmma_f32_16x16x4_f32(
                false, a, false, b, (short)0, acc[nt], false, false);
        }
    }

    // bias (column nt*16+m, uniform over the 8 row-components)
#pragma unroll
    for (int nt = 0; nt < 8; ++nt) {
        float bv = bl[nt * 16 + m];
#pragma unroll
        for (int v = 0; v < 8; ++v) acc[nt][v] += bv;
    }

    // per-row L2 normalize + relu + store
#pragma unroll
    for (int v = 0; v < 8; ++v) {
        float ss = 0.0f;
#pragma unroll
        for (int nt = 0; nt < 8; ++nt) ss += acc[nt][v] * acc[nt][v];
        ss = hadd16(ss);
        const float inv = 1.0f / fmaxf(sqrtf(ss), 1e-12f);
        const int   r   = row0 + v + 8 * half;
#pragma unroll
        for (int nt = 0; nt < 8; ++nt) {
            float val = fmaxf(acc[nt][v] * inv, 0.0f);
            h[(size_t)r * DH + nt * 16 + m] = val;
        }
    }
}

// ------------------------------ layer 2 ------------------------------------
// out = log_softmax(l2norm(mean2 @ Wl2 + b2 + h @ Wr2)), 40 cols padded to 48.
__global__ __launch_bounds__(256) void sage_layer2(
    const float* __restrict__ sum, const float* __restrict__ deg,
    const float* __restrict__ h,
    const float* __restrict__ Wl, const float* __restrict__ bl,
    const float* __restrict__ Wr,
    float* __restrict__ out)
{
    const int wave = threadIdx.x >> 5;
    const int lane = threadIdx.x & 31;
    const int m    = lane & 15;
    const int half = lane >> 4;
    const int row0 = ((int)blockIdx.x * 8 + wave) * 16;
    if (row0 >= NN) return;

    const int   row  = row0 + m;
    const float invd = 1.0f / fmaxf(deg[row], 1.0f);
    const float* aRow = sum + (size_t)row * DH;
    const float* hRow = h   + (size_t)row * DH;

    v8f acc[3] = {};
    bool colok[3];
#pragma unroll
    for (int nt = 0; nt < 3; ++nt) colok[nt] = (nt * 16 + m) < DO;

    // mean2 @ Wl2
    for (int k0 = 0; k0 < DH; k0 += 4) {
        const int k = k0 + half * 2;
        v2f a;
        a.x = aRow[k]     * invd;
        a.y = aRow[k + 1] * invd;
#pragma unroll
        for (int nt = 0; nt < 3; ++nt) {
            const int n = nt * 16 + m;
            v2f b;
            b.x = colok[nt] ? Wl[(size_t)k       * DO + n] : 0.0f;
            b.y = colok[nt] ? Wl[(size_t)(k + 1) * DO + n] : 0.0f;
            acc[nt] = __builtin_amdgcn_wmma_f32_16x16x4_f32(
                false, a, false, b, (short)0, acc[nt], false, false);
        }
    }
    // h @ Wr2
    for (int k0 = 0; k0 < DH; k0 += 4) {
        const int k = k0 + half * 2;
        v2f a;
        a.x = hRow[k];
        a.y = hRow[k + 1];
#pragma unroll
        for (int nt = 0; nt < 3; ++nt) {
            const int n = nt * 16 + m;
            v2f b;
            b.x = colok[nt] ? Wr[(size_t)k       * DO + n] : 0.0f;
            b.y = colok[nt] ? Wr[(size_t)(k + 1) * DO + n] : 0.0f;
            acc[nt] = __builtin_amdgcn_wmma_f32_16x16x4_f32(
                false, a, false, b, (short)0, acc[nt], false, false);
        }
    }

    // bias (padded cols stay exactly 0 -> contribute 0 to sum of squares)
#pragma unroll
    for (int nt = 0; nt < 3; ++nt) {
        float bv = colok[nt] ? bl[nt * 16 + m] : 0.0f;
#pragma unroll
        for (int v = 0; v < 8; ++v) acc[nt][v] += bv;
    }

    // per-row: L2 normalize, then log_softmax over the 40 valid cols
#pragma unroll
    for (int v = 0; v < 8; ++v) {
        float ss = 0.0f;
#pragma unroll
        for (int nt = 0; nt < 3; ++nt) ss += acc[nt][v] * acc[nt][v];
        ss = hadd16(ss);
        const float inv = 1.0f / fmaxf(sqrtf(ss), 1e-12f);

        float vals[3];
        float mx = -INFINITY;
#pragma unroll
        for (int nt = 0; nt < 3; ++nt) {
            vals[nt] = acc[nt][v] * inv;
            if (colok[nt]) mx = fmaxf(mx, vals[nt]);
        }
        mx = hmax16(mx);

        float se = 0.0f;
#pragma unroll
        for (int nt = 0; nt < 3; ++nt)
            if (colok[nt]) se += expf(vals[nt] - mx);
        se = hadd16(se);
        const float lse = logf(se);

        const int r = row0 + v + 8 * half;
#pragma unroll
        for (int nt = 0; nt < 3; ++nt) {
            const int n = nt * 16 + m;
            if (n < DO) out[(size_t)r * DO + n] = vals[nt] - mx - lse;
        }
    }
}

// ------------------------------ launcher -----------------------------------

extern "C" void kernel_launch(void* const* d_in, const int* in_sizes, int n_in,
                              void* d_out, int out_size, void* d_ws, size_t ws_size,
                              hipStream_t stream) {
    const float*     x   = (const float*)d_in[0];
    const long long* ei  = (const long long*)d_in[1];  // int64 edge_index [2, NE]
    const float*     Wl1 = (const float*)d_in[2];
    const float*     bl1 = (const float*)d_in[3];
    const float*     Wr1 = (const float*)d_in[4];
    const float*     Wl2 = (const float*)d_in[5];
    const float*     bl2 = (const float*)d_in[6];
    const float*     Wr2 = (const float*)d_in[7];
    float*           out = (float*)d_out;

    // workspace: sum [NN*DH] | deg [NN] | h [NN*DH]   (~103 MB)
    float* ws  = (float*)d_ws;
    float* sum = ws;
    float* deg = sum + (size_t)NN * DH;
    float* h   = deg + NN;

    const long long* src = ei;
    const long long* dst = ei + NE;

    const int tiles      = NN / 16;              // 6250, NN % 16 == 0
    const int sageBlocks = (tiles + 7) / 8;      // 8 waves (16-row tiles) per block
    const int scatBlocks = (NE + 7) / 8;         // 8 edge-waves per block

    // layer 1: zero sum+deg, scatter x, fused GEMM/norm/relu
    {
        long n = (long)NN * DH + NN;
        zero_kernel<<<(int)((n + 255) / 256), 256, 0, stream>>>(sum, n);
    }
    scatter_kernel<<<scatBlocks, 256, 0, stream>>>(x, src, dst, sum, deg, NE);
    sage_layer1<<<sageBlocks, 256, 0, stream>>>(sum, deg, x, Wl1, bl1, Wr1, h);

    // layer 2: re-zero sum (deg reused), scatter h, fused GEMM/norm/log_softmax
    {
        long n = (long)NN * DH;
        zero_kernel<<<(int)((n + 255) / 256), 256, 0, stream>>>(sum, n);
    }
    scatter_kernel<<<scatBlocks, 256, 0, stream>>>(h, src, dst, sum, nullptr, NE);
    sage_layer2<<<sageBlocks, 256, 0, stream>>>(sum, deg, h, Wl2, bl2, Wr2, out);
}